// TransformerBlock_87247965651443
// MI455X (gfx1250) — compile-verified
//
#include <hip/hip_runtime.h>
#include <hip/hip_bf16.h>
#include <math.h>

typedef float v2f __attribute__((ext_vector_type(2)));
typedef float v8f __attribute__((ext_vector_type(8)));

#define BS    2
#define C     128
#define NL    2
#define NP    8
#define NQ    32768          // level-0 H*W
#define NV    40960
#define NQTOT (BS * NQ)      // 65536 total query rows
#define NRAW  48             // 32 offset cols + 16 attn cols

// Hardcoded level geometry (matches SHAPES in the reference):
//   level 0: H=128, W=256, start=0
//   level 1: H= 64, W=128, start=32768

// ---------------------------------------------------------------------------
// Kernel 1: raw[q][0..47] = query[q] @ [W_off | W_attn]   (M=65536, K=128, N=48)
// WMMA f32 16x16x4, one wave per 16-row M tile, 3 N tiles.
// B staged in LDS *transposed* (n-major, stride 132 == 4 mod 64): each lane's
// (B[k][n], B[k+1][n]) pair is one aligned ds_load_b64, conflict-free.
// ---------------------------------------------------------------------------
#define K1_STRIDE 132
__global__ __launch_bounds__(256) void k_qproj(const float* __restrict__ Q,
                                               const float* __restrict__ Woff,
                                               const float* __restrict__ Wattn,
                                               float* __restrict__ raw) {
    __shared__ float lw[NRAW * K1_STRIDE];     // lw[n][k]
    const int tid = threadIdx.x;
    for (int i = tid; i < 128 * NRAW; i += 256) {
        const int k = i / NRAW, n = i - k * NRAW;   // global-coalesced over n
        lw[n * K1_STRIDE + k] = (n < 32) ? Woff[k * 32 + n]
                                         : Wattn[k * 16 + (n - 32)];
    }
    __syncthreads();

    const int wave = tid >> 5;
    const int lane = tid & 31;
    const int mrow = lane & 15;       // M index within tile (also N lane for B)
    const int half = lane >> 4;       // K half-select per ISA f32 A/B layout
    const long qbase = ((long)blockIdx.x * 8 + wave) * 16;
    const float* aptr = Q + (qbase + mrow) * C + half * 2;
    const float* b0 = &lw[mrow * K1_STRIDE];            // n = mrow
    const float* b1 = &lw[(16 + mrow) * K1_STRIDE];     // n = 16 + mrow
    const float* b2 = &lw[(32 + mrow) * K1_STRIDE];     // n = 32 + mrow

    v8f acc0 = {}, acc1 = {}, acc2 = {};
    for (int kt = 0; kt < 32; ++kt) {
        const int k0 = kt * 4;
        const int krow = k0 + half * 2;
        v2f a  = *(const v2f*)(aptr + k0);   // A[m][k0+2h], A[m][k0+2h+1]
        v2f bb0 = *(const v2f*)(b0 + krow);  // one ds_load_b64 each
        v2f bb1 = *(const v2f*)(b1 + krow);
        v2f bb2 = *(const v2f*)(b2 + krow);
        acc0 = __builtin_amdgcn_wmma_f32_16x16x4_f32(false, a, false, bb0,
                                                     (short)0, acc0, false, false);
        acc1 = __builtin_amdgcn_wmma_f32_16x16x4_f32(false, a, false, bb1,
                                                     (short)0, acc1, false, false);
        acc2 = __builtin_amdgcn_wmma_f32_16x16x4_f32(false, a, false, bb2,
                                                     (short)0, acc2, false, false);
    }
    // D layout: VGPR r -> row (r + 8*half), lane(mrow) -> column
    #pragma unroll
    for (int r = 0; r < 8; ++r) {
        const long row = qbase + r + half * 8;
        raw[row * NRAW + mrow]      = acc0[r];
        raw[row * NRAW + 16 + mrow] = acc1[r];
        raw[row * NRAW + 32 + mrow] = acc2[r];
    }
}

// ---------------------------------------------------------------------------
// Kernel 2: softmax(16) + locations + bilinear gather. One wave per query.
// Lanes 0..15 own the 16 (level,point) pairs (lanes 16..31 mirror them);
// gather phase broadcasts per-point (index, weight) via shuffles while every
// lane accumulates 4 channels (float4). value (42 MB) is L2-resident.
// ---------------------------------------------------------------------------
__global__ __launch_bounds__(256) void k_sample(const float* __restrict__ value,
                                                const float* __restrict__ qloc,
                                                const float* __restrict__ raw,
                                                const float* __restrict__ boff,
                                                const float* __restrict__ battn,
                                                float* __restrict__ sampled) {
    const int lane = threadIdx.x & 31;
    const long q = (long)blockIdx.x * 8 + (threadIdx.x >> 5);
    const int b = (int)(q >> 15);                 // q / NQ (NQ = 2^15)
    const float* rawq = raw + q * NRAW;

    const int pidx = lane & 15;                   // point id 0..15
    const int l    = pidx >> 3;                   // level
    const int Hl = l ? 64 : 128;
    const int Wl = l ? 128 : 256;
    const int lstart = l ? 32768 : 0;

    // softmax over the 16 attention logits (within each 16-lane half)
    float logit = rawq[32 + pidx] + battn[pidx];
    float mx = logit;
    mx = fmaxf(mx, __shfl_xor(mx, 1, 32));
    mx = fmaxf(mx, __shfl_xor(mx, 2, 32));
    mx = fmaxf(mx, __shfl_xor(mx, 4, 32));
    mx = fmaxf(mx, __shfl_xor(mx, 8, 32));
    float e = __expf(logit - mx);
    float s = e;
    s += __shfl_xor(s, 1, 32);
    s += __shfl_xor(s, 2, 32);
    s += __shfl_xor(s, 4, 32);
    s += __shfl_xor(s, 8, 32);
    const float aw = e / s;

    // sampling location (grid_sample, align_corners=False)
    const float offx = rawq[pidx * 2]     + boff[pidx * 2];
    const float offy = rawq[pidx * 2 + 1] + boff[pidx * 2 + 1];
    const float* lp = qloc + (q * NL + l) * 2;
    const float x = (lp[0] + offx / (float)Wl) * (float)Wl - 0.5f;
    const float y = (lp[1] + offy / (float)Hl) * (float)Hl - 0.5f;
    const float xf = floorf(x), yf = floorf(y);
    const float lx = x - xf, ly = y - yf;
    const int x0 = (int)xf, y0 = (int)yf;
    const int x1 = x0 + 1,  y1 = y0 + 1;

    const int cx0 = min(max(x0, 0), Wl - 1), cy0 = min(max(y0, 0), Hl - 1);
    const int cx1 = min(max(x1, 0), Wl - 1), cy1 = min(max(y1, 0), Hl - 1);
    const bool vx0 = (x0 >= 0) & (x0 < Wl), vy0 = (y0 >= 0) & (y0 < Hl);
    const bool vx1 = (x1 >= 0) & (x1 < Wl), vy1 = (y1 >= 0) & (y1 < Hl);

    int   i0 = lstart + cy0 * Wl + cx0;  float w0 = (vx0 & vy0) ? (1.f - lx) * (1.f - ly) * aw : 0.f;
    int   i1 = lstart + cy0 * Wl + cx1;  float w1 = (vx1 & vy0) ? lx * (1.f - ly) * aw : 0.f;
    int   i2 = lstart + cy1 * Wl + cx0;  float w2 = (vx0 & vy1) ? (1.f - lx) * ly * aw : 0.f;
    int   i3 = lstart + cy1 * Wl + cx1;  float w3 = (vx1 & vy1) ? lx * ly * aw : 0.f;

    const float* vb = value + (long)b * NV * C;
    const int c0 = lane * 4;
    float ax = 0.f, ay = 0.f, az = 0.f, awc = 0.f;
    for (int pt = 0; pt < 16; ++pt) {
        const int   j0 = __shfl(i0, pt, 32), j1 = __shfl(i1, pt, 32);
        const int   j2 = __shfl(i2, pt, 32), j3 = __shfl(i3, pt, 32);
        const float g0 = __shfl(w0, pt, 32), g1 = __shfl(w1, pt, 32);
        const float g2 = __shfl(w2, pt, 32), g3 = __shfl(w3, pt, 32);
        const float4 r0 = *(const float4*)(vb + (long)j0 * C + c0);
        const float4 r1 = *(const float4*)(vb + (long)j1 * C + c0);
        const float4 r2 = *(const float4*)(vb + (long)j2 * C + c0);
        const float4 r3 = *(const float4*)(vb + (long)j3 * C + c0);
        ax  += g0 * r0.x + g1 * r1.x + g2 * r2.x + g3 * r3.x;
        ay  += g0 * r0.y + g1 * r1.y + g2 * r2.y + g3 * r3.y;
        az  += g0 * r0.z + g1 * r1.z + g2 * r2.z + g3 * r3.z;
        awc += g0 * r0.w + g1 * r1.w + g2 * r2.w + g3 * r3.w;
    }
    float4 o; o.x = ax; o.y = ay; o.z = az; o.w = awc;
    *(float4*)(sampled + q * C + c0) = o;
}

// ---------------------------------------------------------------------------
// Kernel 3: out = sampled @ W_out + b_out   (M=65536, K=128, N=128)
// WMMA f32 16x16x4; W_out staged transposed in LDS in two 64-row K halves
// (n-major, stride 68 == 4 mod 64 -> conflict-free b64 reads; 34.8 KB).
// ---------------------------------------------------------------------------
#define K3_STRIDE 68
__global__ __launch_bounds__(256) void k_outproj(const float* __restrict__ A,
                                                 const float* __restrict__ Wout,
                                                 const float* __restrict__ bout,
                                                 float* __restrict__ out) {
    __shared__ float lw[128 * K3_STRIDE];   // lw[n][kk], kk in 0..63
    const int tid = threadIdx.x;
    const int wave = tid >> 5;
    const int lane = tid & 31;
    const int mrow = lane & 15;
    const int half = lane >> 4;
    const long qbase = ((long)blockIdx.x * 8 + wave) * 16;
    const float* aptr = A + (qbase + mrow) * C + half * 2;

    v8f acc[8];
    #pragma unroll
    for (int i = 0; i < 8; ++i) acc[i] = (v8f){};

    for (int kh = 0; kh < 2; ++kh) {
        __syncthreads();      // guard previous-half reads before overwrite
        for (int i = tid; i < 64 * 128; i += 256) {
            const int k = i >> 7, n = i & 127;   // global-coalesced over n
            lw[n * K3_STRIDE + k] = Wout[(kh * 64 + k) * 128 + n];
        }
        __syncthreads();
        for (int kt = 0; kt < 16; ++kt) {
            const int k0 = kh * 64 + kt * 4;
            const int krow = kt * 4 + half * 2;         // row within this half
            v2f a = *(const v2f*)(aptr + k0);
            #pragma unroll
            for (int nt = 0; nt < 8; ++nt) {
                const int n = nt * 16 + mrow;
                v2f bb = *(const v2f*)(&lw[n * K3_STRIDE + krow]);  // ds_load_b64
                acc[nt] = __builtin_amdgcn_wmma_f32_16x16x4_f32(false, a, false, bb,
                                                                (short)0, acc[nt],
                                                                false, false);
            }
        }
    }

    #pragma unroll
    for (int nt = 0; nt < 8; ++nt) {
        const int n = nt * 16 + mrow;
        const float bias = bout[n];
        #pragma unroll
        for (int r = 0; r < 8; ++r) {
            const long row = qbase + r + half * 8;
            out[row * C + n] = acc[nt][r] + bias;
        }
    }
}

// ---------------------------------------------------------------------------
extern "C" void kernel_launch(void* const* d_in, const int* in_sizes, int n_in,
                              void* d_out, int out_size, void* d_ws, size_t ws_size,
                              hipStream_t stream) {
    (void)in_sizes; (void)n_in; (void)out_size; (void)ws_size;
    const float* query  = (const float*)d_in[0];   // [2, 32768, 128]
    const float* value  = (const float*)d_in[1];   // [2, 40960, 128]
    const float* qloc   = (const float*)d_in[2];   // [2, 32768, 2, 2]
    // d_in[3] spatial_shapes (int64), d_in[4] level_start_index (int64): fixed, hardcoded
    const float* W_off  = (const float*)d_in[5];   // [128, 32]
    const float* b_off  = (const float*)d_in[6];   // [32]
    const float* W_attn = (const float*)d_in[7];   // [128, 16]
    const float* b_attn = (const float*)d_in[8];   // [16]
    const float* W_out  = (const float*)d_in[9];   // [128, 128]
    const float* b_out  = (const float*)d_in[10];  // [128]
    float* out = (float*)d_out;                    // [2, 32768, 128]

    float* raw     = (float*)d_ws;                     // 65536 * 48 f32 (12.6 MB)
    float* sampled = raw + (size_t)NQTOT * NRAW;       // 65536 * 128 f32 (33.6 MB)

    k_qproj  <<<NQTOT / 128, 256, 0, stream>>>(query, W_off, W_attn, raw);
    k_sample <<<NQTOT / 8,   256, 0, stream>>>(value, qloc, raw, b_off, b_attn, sampled);
    k_outproj<<<NQTOT / 128, 256, 0, stream>>>(sampled, W_out, b_out, out);
}